// GraphDecoderHomo_62981400429114
// MI455X (gfx1250) — compile-verified
//
#include <hip/hip_runtime.h>
#include <hip/hip_bf16.h>

typedef __attribute__((ext_vector_type(2))) float v2f;
typedef __attribute__((ext_vector_type(8))) float v8f;

// ---------------------------------------------------------------------------
// Utility: zero a float buffer
// ---------------------------------------------------------------------------
__global__ __launch_bounds__(256) void k_zero_f32(float* __restrict__ p, int n) {
    int i = blockIdx.x * blockDim.x + threadIdx.x;
    if (i < n) p[i] = 0.0f;
}

// ---------------------------------------------------------------------------
// Degree accumulation (unweighted, per reference) via f32 atomics
// ---------------------------------------------------------------------------
__global__ __launch_bounds__(256) void k_degrees(const int* __restrict__ src,
                                                 const int* __restrict__ dst,
                                                 float* __restrict__ outdeg,
                                                 float* __restrict__ indeg,
                                                 int E) {
    int e = blockIdx.x * blockDim.x + threadIdx.x;
    if (e >= E) return;
    atomicAdd(&outdeg[src[e]], 1.0f);
    atomicAdd(&indeg[dst[e]], 1.0f);
}

// deg -> clip(deg,1)^-0.5, in place (covers both norm buffers at once)
__global__ __launch_bounds__(256) void k_deg_to_norm(float* __restrict__ p, int n) {
    int i = blockIdx.x * blockDim.x + threadIdx.x;
    if (i < n) p[i] = rsqrtf(fmaxf(p[i], 1.0f));
}

// ---------------------------------------------------------------------------
// H = (X * norm_src[:,None]) @ W    using V_WMMA_F32_16X16X4_F32
//   X: [n, DIN], W: [DIN, DOUT], H: [n, DOUT]
//
// Block = one 16-row M-tile; wave w = N-tile w (blockDim = (DOUT/16)*32).
// The scaled A-tile is staged once into LDS (padded rows: bank-conflict-free
// ds_load_b64 for the 16 lanes reading one column pair), then every wave runs
// the K loop from LDS while streaming B from global (W is cache-resident).
//
// A (16x4 f32, 2 VGPRs/lane): lanes 0-15 -> K = k,k+1 ; lanes 16-31 -> K = k+2,k+3
// B (4x16 f32, 2 VGPRs/lane): VGPR0 row = k+2*hi, VGPR1 row = k+2*hi+1, col = lane&15
// C/D (16x16 f32, 8 VGPRs):   VGPR j -> row (j + 8*hi), col = lane&15
// ---------------------------------------------------------------------------
template <int DIN, int DOUT>
__global__ __launch_bounds__((DOUT / 16) * 32) void
k_gemm_norm_wmma(const float* __restrict__ X,
                 const float* __restrict__ norm_src,
                 const float* __restrict__ W,
                 float* __restrict__ H) {
    constexpr int NT     = (DOUT / 16) * 32;  // threads per block
    constexpr int STRIDE = DIN + 4;           // padded LDS row (bank shift of 4)
    __shared__ float As[16 * STRIDE];

    const int tid  = threadIdx.x;
    const int lane = tid & 31;
    const int wave = tid >> 5;                // == N-tile index
    const int l15  = lane & 15;
    const int hi   = lane >> 4;               // 0 or 1
    const int row0 = blockIdx.x * 16;

    // --- stage scaled A-tile: 16 x DIN floats as float4s ----------------
    constexpr int NV4 = 16 * DIN / 4;
    for (int v = tid; v < NV4; v += NT) {
        const int r  = v / (DIN / 4);
        const int c4 = (v - r * (DIN / 4)) * 4;
        float4 x = *(const float4*)(X + (size_t)(row0 + r) * DIN + c4);
        const float ns = norm_src[row0 + r];
        x.x *= ns; x.y *= ns; x.z *= ns; x.w *= ns;
        *(float4*)(&As[r * STRIDE + c4]) = x;
    }
    __syncthreads();

    const int col0 = wave * 16;
    const float* __restrict__ wcol = W + col0 + l15;
    const float* __restrict__ arow = &As[l15 * STRIDE + 2 * hi];

    v8f c = {};
    for (int k = 0; k < DIN; k += 4) {
        v2f a = *(const v2f*)(arow + k);      // ds_load_b64, conflict-free
        const int kr = k + 2 * hi;
        v2f b;
        b.x = wcol[(size_t)kr * DOUT];
        b.y = wcol[(size_t)(kr + 1) * DOUT];
        c = __builtin_amdgcn_wmma_f32_16x16x4_f32(
                /*neg_a=*/false, a, /*neg_b=*/false, b,
                /*c_mod=*/(short)0, c, /*reuse_a=*/false, /*reuse_b=*/false);
    }

    float* __restrict__ hrow = H + (size_t)(row0 + 8 * hi) * DOUT + col0 + l15;
#pragma unroll
    for (int j = 0; j < 8; ++j) hrow[(size_t)j * DOUT] = c[j];
}

// ---------------------------------------------------------------------------
// Edge scatter: AGG[dst[e], :] += H[src[e], :] * ew[e]
// One wave per edge; each lane owns DOUT/32 contiguous channels.
// ---------------------------------------------------------------------------
template <int DOUT>
__global__ __launch_bounds__(256) void k_scatter_edges(const float* __restrict__ H,
                                                       const int* __restrict__ src,
                                                       const int* __restrict__ dst,
                                                       const float* __restrict__ ew,
                                                       float* __restrict__ AGG,
                                                       int E) {
    const int lane = threadIdx.x & 31;
    const int e    = (blockIdx.x * blockDim.x + threadIdx.x) >> 5;
    if (e >= E) return;
    const int   s = src[e];
    const int   d = dst[e];
    const float w = ew[e];
    constexpr int PER = DOUT / 32;            // 4 (dout=128) or 2 (dout=64)
    const float* __restrict__ hp = H + (size_t)s * DOUT + lane * PER;
    float* __restrict__       ap = AGG + (size_t)d * DOUT + lane * PER;
    if constexpr (PER == 4) {
        float4 v = *(const float4*)hp;
        atomicAdd(ap + 0, v.x * w);
        atomicAdd(ap + 1, v.y * w);
        atomicAdd(ap + 2, v.z * w);
        atomicAdd(ap + 3, v.w * w);
    } else {
        float2 v = *(const float2*)hp;
        atomicAdd(ap + 0, v.x * w);
        atomicAdd(ap + 1, v.y * w);
    }
}

// ---------------------------------------------------------------------------
// out = act(AGG * norm_dst[node] + bias[ch]), in place
// ---------------------------------------------------------------------------
template <int DOUT, bool TANH>
__global__ __launch_bounds__(256) void k_finalize(float* __restrict__ AGG,
                                                  const float* __restrict__ norm_dst,
                                                  const float* __restrict__ bias,
                                                  int total) {
    int i = blockIdx.x * blockDim.x + threadIdx.x;
    if (i >= total) return;
    const int node = i / DOUT;
    const int ch   = i - node * DOUT;
    float v = AGG[i] * norm_dst[node] + bias[ch];
    AGG[i] = TANH ? tanhf(v) : v;
}

// ---------------------------------------------------------------------------
// Layer 3 projection (64 -> 1): H3[n] = dot(X3[n,:], W3[:,0]) * norm_src[n]
// ---------------------------------------------------------------------------
__global__ __launch_bounds__(256) void k_dot64_norm(const float* __restrict__ X,
                                                    const float* __restrict__ norm_src,
                                                    const float* __restrict__ W3,
                                                    float* __restrict__ H3,
                                                    int n) {
    int i = blockIdx.x * blockDim.x + threadIdx.x;
    if (i >= n) return;
    const float* __restrict__ x = X + (size_t)i * 64;
    float acc = 0.0f;
#pragma unroll
    for (int k = 0; k < 64; ++k) acc = fmaf(x[k], W3[k], acc);
    H3[i] = acc * norm_src[i];
}

// Layer 3 edge scatter (scalar channel)
__global__ __launch_bounds__(256) void k_scatter_edges1(const float* __restrict__ H3,
                                                        const int* __restrict__ src,
                                                        const int* __restrict__ dst,
                                                        const float* __restrict__ ew,
                                                        float* __restrict__ AGG,
                                                        int E) {
    int e = blockIdx.x * blockDim.x + threadIdx.x;
    if (e >= E) return;
    atomicAdd(&AGG[dst[e]], H3[src[e]] * ew[e]);
}

// ---------------------------------------------------------------------------
// Host-side orchestration
// ---------------------------------------------------------------------------
static inline int cdiv(int a, int b) { return (a + b - 1) / b; }

extern "C" void kernel_launch(void* const* d_in, const int* in_sizes, int n_in,
                              void* d_out, int out_size, void* d_ws, size_t ws_size,
                              hipStream_t stream) {
    const float* b_z = (const float*)d_in[0];
    const int*   src = (const int*)d_in[1];
    const int*   dst = (const int*)d_in[2];
    const float* ew  = (const float*)d_in[3];
    const float* W1  = (const float*)d_in[4];
    const float* b1  = (const float*)d_in[5];
    const float* W2  = (const float*)d_in[6];
    const float* b2  = (const float*)d_in[7];
    const float* W3  = (const float*)d_in[8];
    const float* b3  = (const float*)d_in[9];

    const int n = in_sizes[0] / 256;   // 32768 nodes (D = 256)
    const int E = in_sizes[1];         // 524288 edges

    // Workspace layout (floats)
    float* outnorm = (float*)d_ws;                 // [n]
    float* innorm  = outnorm + n;                  // [n]
    float* H1      = innorm + n;                   // [n,128]
    float* X2      = H1 + (size_t)n * 128;         // [n,128] (AGG1 -> activations)
    float* H2      = X2 + (size_t)n * 128;         // [n,64]
    float* X3      = H2 + (size_t)n * 64;          // [n,64]  (AGG2 -> activations)
    float* H3      = X3 + (size_t)n * 64;          // [n]
    float* out     = (float*)d_out;                // [n]

    const int TB = 256;

    // --- degree norms ---------------------------------------------------
    k_zero_f32<<<cdiv(2 * n, TB), TB, 0, stream>>>(outnorm, 2 * n);
    k_degrees<<<cdiv(E, TB), TB, 0, stream>>>(src, dst, outnorm, innorm, E);
    k_deg_to_norm<<<cdiv(2 * n, TB), TB, 0, stream>>>(outnorm, 2 * n);

    // --- layer 1: 256 -> 128, tanh --------------------------------------
    {
        k_gemm_norm_wmma<256, 128><<<n / 16, (128 / 16) * 32, 0, stream>>>(b_z, outnorm, W1, H1);
        k_zero_f32<<<cdiv(n * 128, TB), TB, 0, stream>>>(X2, n * 128);
        k_scatter_edges<128><<<cdiv(E * 32, TB), TB, 0, stream>>>(H1, src, dst, ew, X2, E);
        k_finalize<128, true><<<cdiv(n * 128, TB), TB, 0, stream>>>(X2, innorm, b1, n * 128);
    }

    // --- layer 2: 128 -> 64, tanh ---------------------------------------
    {
        k_gemm_norm_wmma<128, 64><<<n / 16, (64 / 16) * 32, 0, stream>>>(X2, outnorm, W2, H2);
        k_zero_f32<<<cdiv(n * 64, TB), TB, 0, stream>>>(X3, n * 64);
        k_scatter_edges<64><<<cdiv(E * 32, TB), TB, 0, stream>>>(H2, src, dst, ew, X3, E);
        k_finalize<64, true><<<cdiv(n * 64, TB), TB, 0, stream>>>(X3, innorm, b2, n * 64);
    }

    // --- layer 3: 64 -> 1, linear, straight into d_out -------------------
    {
        k_dot64_norm<<<cdiv(n, TB), TB, 0, stream>>>(X3, outnorm, W3, H3, n);
        k_zero_f32<<<cdiv(n, TB), TB, 0, stream>>>(out, n);
        k_scatter_edges1<<<cdiv(E, TB), TB, 0, stream>>>(H3, src, dst, ew, out, E);
        k_finalize<1, false><<<cdiv(n, TB), TB, 0, stream>>>(out, innorm, b3, n);
    }
}